// GAT_Layer_3985729650941
// MI455X (gfx1250) — compile-verified
//
#include <hip/hip_runtime.h>
#include <math.h>

typedef float v2f __attribute__((ext_vector_type(2)));
typedef float v8f __attribute__((ext_vector_type(8)));

#define DIM 64
#define NEG_ATT 0.2f
#define NEG_OUT 0.01f

// ---------------- fill ----------------
__global__ void fill_kernel(float* __restrict__ p, float v, int n) {
    int i = blockIdx.x * blockDim.x + threadIdx.x;
    if (i < n) p[i] = v;
}

// ---------------- XW = X[N,64] @ W[64,64], fp32 WMMA ----------------
// One wave computes one 16x16 output tile. K stepped by 4 via V_WMMA_F32_16X16X4_F32.
__global__ __launch_bounds__(256) void gemm64_wmma(const float* __restrict__ X,
                                                   const float* __restrict__ W,
                                                   float* __restrict__ XW,
                                                   int n_row_tiles) {
    int wave   = blockIdx.x * (blockDim.x >> 5) + (threadIdx.x >> 5);
    int tile_m = wave >> 2;        // 4 column tiles (64/16)
    int tile_n = wave & 3;
    if (tile_m >= n_row_tiles) return;

    int lane = threadIdx.x & 31;
    int half = lane >> 4;          // 0: K pair {0,1}; 1: K pair {2,3}
    int l    = lane & 15;          // M index for A, N index for B

    const float* xrow = X + (size_t)(tile_m * 16 + l) * DIM;   // A row
    const float* wcol = W + tile_n * 16 + l;                   // B column

    v8f c = {0.f, 0.f, 0.f, 0.f, 0.f, 0.f, 0.f, 0.f};
#pragma unroll
    for (int k = 0; k < DIM; k += 4) {
        v2f a, b;
        a.x = xrow[k + 2 * half];
        a.y = xrow[k + 2 * half + 1];
        b.x = wcol[(size_t)(k + 2 * half) * DIM];
        b.y = wcol[(size_t)(k + 2 * half + 1) * DIM];
        c = __builtin_amdgcn_wmma_f32_16x16x4_f32(false, a, false, b,
                                                  (short)0, c, false, false);
    }
    // D layout: VGPR i holds row (i + 8*half), lanes 0-15 / 16-31 are N=0..15
    float* orow = XW + (size_t)(tile_m * 16 + 8 * half) * DIM + tile_n * 16 + l;
#pragma unroll
    for (int i = 0; i < 8; ++i) orow[(size_t)i * DIM] = c[i];
}

// ---------------- a_src[n] = xw[n].att_src ; a_dst likewise ----------------
__global__ __launch_bounds__(256) void attn_dots(const float* __restrict__ XW,
                                                 const float* __restrict__ as,
                                                 const float* __restrict__ ad,
                                                 float* __restrict__ a_src,
                                                 float* __restrict__ a_dst, int n) {
    int node = (blockIdx.x * blockDim.x + threadIdx.x) >> 5;
    int lane = threadIdx.x & 31;
    if (node >= n) return;
    const float* row = XW + (size_t)node * DIM;
    float vs = row[lane] * as[lane] + row[lane + 32] * as[lane + 32];
    float vd = row[lane] * ad[lane] + row[lane + 32] * ad[lane + 32];
#pragma unroll
    for (int off = 16; off > 0; off >>= 1) {
        vs += __shfl_xor(vs, off, 32);
        vd += __shfl_xor(vd, off, 32);
    }
    if (lane == 0) { a_src[node] = vs; a_dst[node] = vd; }
}

// ---------------- edge pass 1: alpha + segment max ----------------
__global__ void edge_alpha_max(const int* __restrict__ esrc, const int* __restrict__ edst,
                               const float* __restrict__ a_src, const float* __restrict__ a_dst,
                               float* __restrict__ alpha, float* __restrict__ amax,
                               int E, int Etot) {
    int e = blockIdx.x * blockDim.x + threadIdx.x;
    if (e >= Etot) return;
    int s, d;
    if (e < E) { s = esrc[e]; d = edst[e]; } else { s = e - E; d = s; }
    float a = a_src[s] + a_dst[d];
    a = (a > 0.f) ? a : NEG_ATT * a;
    alpha[e] = a;
    atomicMax(amax + d, a);   // global_atomic_max_num_f32
}

// ---------------- edge pass 2: exp + segment sum ----------------
__global__ void edge_exp_sum(const int* __restrict__ edst,
                             float* __restrict__ alpha, const float* __restrict__ amax,
                             float* __restrict__ denom, int E, int Etot) {
    int e = blockIdx.x * blockDim.x + threadIdx.x;
    if (e >= Etot) return;
    int d = (e < E) ? edst[e] : (e - E);
    float ex = __expf(alpha[e] - amax[d]);
    alpha[e] = ex;                 // reuse buffer as ex
    atomicAdd(denom + d, ex);      // global_atomic_add_f32
}

// ---------------- edge pass 3: weighted scatter ----------------
// 16 threads per edge, 4 channels (float4) each.
__global__ void edge_scatter(const int* __restrict__ esrc, const int* __restrict__ edst,
                             const float* __restrict__ XW, const float* __restrict__ ex,
                             const float* __restrict__ denom, float* __restrict__ out,
                             int E, int Etot) {
    int t = blockIdx.x * blockDim.x + threadIdx.x;
    int e = t >> 4;
    if (e >= Etot) return;
    int g = t & 15;
    int s, d;
    if (e < E) { s = esrc[e]; d = edst[e]; } else { s = e - E; d = s; }
    float coef = ex[e] / (denom[d] + 1e-16f);
    const float4 x4 = *(const float4*)(XW + (size_t)s * DIM + g * 4);
    float* o = out + (size_t)d * DIM + g * 4;
    atomicAdd(o + 0, x4.x * coef);
    atomicAdd(o + 1, x4.y * coef);
    atomicAdd(o + 2, x4.z * coef);
    atomicAdd(o + 3, x4.w * coef);
}

// ---------------- finalize: out = leaky_relu(out + b, 0.01) ----------------
__global__ void bias_lrelu(float* __restrict__ out, const float* __restrict__ b, int total) {
    int i = blockIdx.x * blockDim.x + threadIdx.x;
    if (i >= total) return;
    float v = out[i] + b[i & (DIM - 1)];
    out[i] = (v > 0.f) ? v : NEG_OUT * v;
}

// ---------------- host side ----------------
static inline int cdiv(int a, int b) { return (a + b - 1) / b; }

static void run_layer(const float* x, const float* W, const float* att_s, const float* att_d,
                      const float* bias, const int* esrc, const int* edst,
                      int N, int E, int Etot,
                      float* xw, float* a_src, float* a_dst, float* amax, float* denom,
                      float* alpha, float* outbuf, hipStream_t stream) {
    // 1) projection via WMMA
    int n_row_tiles = N / 16;                       // N = 100000 is a multiple of 16
    int waves = n_row_tiles * 4;
    gemm64_wmma<<<cdiv(waves, 8), 256, 0, stream>>>(x, W, xw, n_row_tiles);

    // 2) attention dot products
    attn_dots<<<cdiv(N * 32, 256), 256, 0, stream>>>(xw, att_s, att_d, a_src, a_dst, N);

    // 3) init per-node stats and output accumulator
    fill_kernel<<<cdiv(N, 256), 256, 0, stream>>>(amax, -INFINITY, N);
    fill_kernel<<<cdiv(N, 256), 256, 0, stream>>>(denom, 0.f, N);
    fill_kernel<<<cdiv(N * DIM, 256), 256, 0, stream>>>(outbuf, 0.f, N * DIM);

    // 4) segment softmax + scatter
    edge_alpha_max<<<cdiv(Etot, 256), 256, 0, stream>>>(esrc, edst, a_src, a_dst,
                                                        alpha, amax, E, Etot);
    edge_exp_sum<<<cdiv(Etot, 256), 256, 0, stream>>>(edst, alpha, amax, denom, E, Etot);
    edge_scatter<<<cdiv(Etot * 16, 256), 256, 0, stream>>>(esrc, edst, xw, alpha, denom,
                                                           outbuf, E, Etot);

    // 5) bias + output leaky relu
    bias_lrelu<<<cdiv(N * DIM, 256), 256, 0, stream>>>(outbuf, bias, N * DIM);
}

extern "C" void kernel_launch(void* const* d_in, const int* in_sizes, int n_in,
                              void* d_out, int out_size, void* d_ws, size_t ws_size,
                              hipStream_t stream) {
    const int*   edge_index = (const int*)d_in[0];   // [2, E]
    const float* emb        = (const float*)d_in[1]; // [N, 64]
    const float* W1         = (const float*)d_in[2];
    const float* att_src1   = (const float*)d_in[3];
    const float* att_dst1   = (const float*)d_in[4];
    const float* b1         = (const float*)d_in[5];
    const float* W2         = (const float*)d_in[6];
    const float* att_src2   = (const float*)d_in[7];
    const float* att_dst2   = (const float*)d_in[8];
    const float* b2         = (const float*)d_in[9];

    const int E    = in_sizes[0] / 2;
    const int N    = in_sizes[1] / DIM;
    const int Etot = E + N;                          // self-loops appended

    const int* esrc = edge_index;
    const int* edst = edge_index + E;

    float* ws     = (float*)d_ws;
    float* xw     = ws;                               // N*DIM
    float* h1     = xw + (size_t)N * DIM;             // N*DIM
    float* a_src  = h1 + (size_t)N * DIM;             // N
    float* a_dst  = a_src + N;                        // N
    float* amax   = a_dst + N;                        // N
    float* denom  = amax + N;                         // N
    float* alpha  = denom + N;                        // Etot

    // layer 1: emb -> h1
    run_layer(emb, W1, att_src1, att_dst1, b1, esrc, edst, N, E, Etot,
              xw, a_src, a_dst, amax, denom, alpha, h1, stream);
    // layer 2: h1 -> d_out
    run_layer(h1, W2, att_src2, att_dst2, b2, esrc, edst, N, E, Etot,
              xw, a_src, a_dst, amax, denom, alpha, (float*)d_out, stream);
}